// Relation_Refine_26637387170178
// MI455X (gfx1250) — compile-verified
//
#include <hip/hip_runtime.h>
#include <math.h>

typedef float v2f __attribute__((ext_vector_type(2)));
typedef float v8f __attribute__((ext_vector_type(8)));

#define G1N   8
#define C1N   128
#define C2N   64
#define NB    256      // 16x16 spatial blocks
#define BATCH 4
#define PIX1  16384    // 128*128
#define PIX2  65536    // 256*256
#define BN_EPS 1e-5f

// ---------------------------------------------------------------------------
// K1: per-(b,c) plane means for gap1 [4*128] and gap2 [4*64]
// ---------------------------------------------------------------------------
__global__ void k_plane_mean(const float* __restrict__ x1,
                             const float* __restrict__ x2,
                             float* __restrict__ gap1,
                             float* __restrict__ gap2) {
    __shared__ float red[256];
    int tid = threadIdx.x, bid = blockIdx.x;
    const float* src; int len; float* dst; float inv;
    if (bid < BATCH * C1N) {
        src = x1 + (size_t)bid * PIX1; len = PIX1; dst = gap1 + bid; inv = 1.0f / PIX1;
    } else {
        int i2 = bid - BATCH * C1N;
        src = x2 + (size_t)i2 * PIX2; len = PIX2; dst = gap2 + i2; inv = 1.0f / PIX2;
    }
    float acc = 0.f;
    for (int i = tid; i < len; i += 256) acc += src[i];
    red[tid] = acc; __syncthreads();
    for (int s = 128; s > 0; s >>= 1) {
        if (tid < s) red[tid] += red[tid + s];
        __syncthreads();
    }
    if (tid == 0) *dst = red[0] * inv;
}

// ---------------------------------------------------------------------------
// K2: channel attention -> cw [4*64]  (one 256-thread block; tid = b*64+g*8+o)
// ---------------------------------------------------------------------------
__global__ void k_channel_attn(const float* __restrict__ gap1, const float* __restrict__ gap2,
                               const float* __restrict__ w_c1, const float* __restrict__ b_c1,
                               const float* __restrict__ w_c2, const float* __restrict__ b_c2,
                               float* __restrict__ cw) {
    __shared__ float m2[256];
    __shared__ float lg[256];
    int tid = threadIdx.x;
    int b = tid >> 6, g = (tid >> 3) & 7, o = tid & 7;
    float a1 = b_c1[o];
    const float* g1 = gap1 + b * C1N + g * 16;
    for (int c = 0; c < 16; ++c) a1 += w_c1[o * 16 + c] * g1[c];
    a1 = fmaxf(a1, 0.f);
    float a2 = b_c2[o];
    const float* g2 = gap2 + b * C2N + g * 8;
    for (int c = 0; c < 8; ++c) a2 += w_c2[o * 8 + c] * g2[c];
    a2 = fmaxf(a2, 0.f);
    m2[tid] = a2;
    __syncthreads();
    int base = tid & ~7;
    float s2 = 0.f;
    for (int p = 0; p < 8; ++p) { float v = m2[base + p]; s2 += v * v; }
    lg[tid] = a1 * s2;
    __syncthreads();
    float mx = -1e30f;
    for (int p = 0; p < 8; ++p) mx = fmaxf(mx, lg[base + p]);
    float den = 0.f;
    for (int p = 0; p < 8; ++p) den += __expf(lg[base + p] - mx);
    cw[tid] = __expf(lg[tid] - mx) / den;   // tid == b*64 + channel
}

// ---------------------------------------------------------------------------
// K3: pos2 [4][256 blocks][256] in block-major layout (streams x2 once)
// ---------------------------------------------------------------------------
__global__ void k_pos2(const float* __restrict__ x2, const float* __restrict__ w_p2,
                       const float* __restrict__ b_p2, float* __restrict__ pos2) {
    __shared__ float w[C2N];
    int tid = threadIdx.x;
    if (tid < C2N) w[tid] = w_p2[tid];
    __syncthreads();
    int p  = blockIdx.x * 256 + tid;
    int b  = p >> 16;
    int pp = p & 0xFFFF;
    const float* src = x2 + (size_t)b * C2N * PIX2 + pp;
    float acc = b_p2[0];
    #pragma unroll 8
    for (int c = 0; c < C2N; ++c) acc += w[c] * src[(size_t)c * PIX2];
    int h = pp >> 8, ww = pp & 255;
    int n  = (h >> 4) * 16 + (ww >> 4);
    int ij = (h & 15) * 16 + (ww & 15);
    pos2[((b * NB + n) << 8) + ij] = acc;
}

// ---------------------------------------------------------------------------
// K4: pos1 [4][256 blocks][64] in block-major layout (streams x1 once)
// ---------------------------------------------------------------------------
__global__ void k_pos1(const float* __restrict__ x1, const float* __restrict__ w_p1,
                       const float* __restrict__ b_p1, float* __restrict__ pos1) {
    __shared__ float w[C1N];
    int tid = threadIdx.x;
    if (tid < C1N) w[tid] = w_p1[tid];
    __syncthreads();
    int p  = blockIdx.x * 256 + tid;
    int b  = p >> 14;          // /16384
    int pp = p & 0x3FFF;
    const float* src = x1 + (size_t)b * C1N * PIX1 + pp;
    float acc = b_p1[0];
    #pragma unroll 8
    for (int c = 0; c < C1N; ++c) acc += w[c] * src[(size_t)c * PIX1];
    int h = pp >> 7, ww = pp & 127;
    int n  = (h >> 3) * 16 + (ww >> 3);
    int ij = (h & 7) * 8 + (ww & 7);
    pos1[((b * NB + n) << 6) + ij] = acc;
}

// ---------------------------------------------------------------------------
// K5: x1t = w_lin[256x64] @ pos1^T + b_lin, via V_WMMA_F32_16X16X4_F32
//     GEMM: M=256 (o), K=64 (c), N=1024 (b*nb). Wave = one 16x16 tile.
// ---------------------------------------------------------------------------
__global__ void k_x1t_wmma(const float* __restrict__ w_lin, const float* __restrict__ b_lin,
                           const float* __restrict__ pos1, float* __restrict__ x1t) {
    int tid = threadIdx.x;
    int lane = tid & 31, wv = tid >> 5;
    int wid = blockIdx.x * 8 + wv;     // 0..1023
    int mtile = wid >> 6;              // 0..15
    int ntile = wid & 63;              // 0..63
    int half = lane >> 4, ln = lane & 15;
    int m0 = mtile * 16, n0 = ntile * 16;
    const float* arow = w_lin + (m0 + ln) * 64;   // A: row m = ln
    const float* brow = pos1 + (n0 + ln) * 64;    // B: column n = ln (pos1 row-major [bn][c])
    v8f c = {};
    #pragma unroll
    for (int k4 = 0; k4 < 16; ++k4) {
        int kk = k4 * 4 + (half ? 2 : 0);
        v2f a, bb;
        a.x  = arow[kk];  a.y  = arow[kk + 1];
        bb.x = brow[kk];  bb.y = brow[kk + 1];
        c = __builtin_amdgcn_wmma_f32_16x16x4_f32(false, a, false, bb, (short)0, c, false, false);
    }
    int col = n0 + ln;                 // bn index
    #pragma unroll
    for (int i = 0; i < 8; ++i) {
        int o = m0 + i + half * 8;
        x1t[col * 256 + o] = c[i] + b_lin[o];
    }
}

// ---------------------------------------------------------------------------
// K6: per-block rank-1 softmax attention:
//     weighted[k] = sum_q pos2[q] * exp(pos2[q]*x1t[k] - m_q) / denom[q]
//     One 256-thread block per (b,n); x1t/pos2 cached in LDS.
// ---------------------------------------------------------------------------
__global__ void k_spa(const float* __restrict__ pos2, const float* __restrict__ x1t,
                      float* __restrict__ spa) {
    __shared__ float xk[256], p2[256], coef[256], mq[256], red[256];
    int tid = threadIdx.x;
    int bn = blockIdx.x;               // b*256 + n
    float x = x1t[bn * 256 + tid];
    float s = pos2[bn * 256 + tid];
    xk[tid] = x; p2[tid] = s;
    red[tid] = x; __syncthreads();
    for (int st = 128; st > 0; st >>= 1) {
        if (tid < st) red[tid] = fmaxf(red[tid], red[tid + st]);
        __syncthreads();
    }
    float maxX = red[0]; __syncthreads();
    red[tid] = x; __syncthreads();
    for (int st = 128; st > 0; st >>= 1) {
        if (tid < st) red[tid] = fminf(red[tid], red[tid + st]);
        __syncthreads();
    }
    float minX = red[0]; __syncthreads();
    // phase 1: this thread owns row q = tid
    float m = (s >= 0.f) ? s * maxX : s * minX;   // row max of s*x1t[:]
    float d = 0.f;
    for (int k = 0; k < 256; ++k) d += __expf(s * xk[k] - m);
    coef[tid] = s / d;
    mq[tid] = m;
    __syncthreads();
    // phase 2: this thread owns column k = tid
    float acc = 0.f;
    for (int q = 0; q < 256; ++q) acc += coef[q] * __expf(p2[q] * x - mq[q]);
    spa[bn * 256 + tid] = acc;         // raw flatten: == spa[b, n*256 + k] image row n
}

// ---------------------------------------------------------------------------
// Kprep: W'[b,o,c] = w_f[o,c]*cw[b,c]; BN folded: sA=gamma/sqrt(1+eps),
//        sR[o] = (sum_c w_f[o,c]) * sA[o], sB[o] = b_f[o]*sA[o] + beta[o]
// ---------------------------------------------------------------------------
__global__ void k_prep(const float* __restrict__ w_f, const float* __restrict__ b_f,
                       const float* __restrict__ gamma, const float* __restrict__ beta,
                       const float* __restrict__ cw,
                       float* __restrict__ Wp, float* __restrict__ sA,
                       float* __restrict__ sR, float* __restrict__ sB) {
    int idx = blockIdx.x * 256 + threadIdx.x;   // 0..16383
    int b = idx >> 12, oc = idx & 4095, ccc = oc & 63;
    Wp[idx] = w_f[oc] * cw[b * 64 + ccc];
    if (blockIdx.x == 0 && threadIdx.x < 64) {
        int o = threadIdx.x;
        float r = 0.f;
        for (int c = 0; c < 64; ++c) r += w_f[o * 64 + c];
        float A = gamma[o] * rsqrtf(1.0f + BN_EPS);
        sA[o] = A; sR[o] = r * A; sB[o] = b_f[o] * A + beta[o];
    }
}

// ---------------------------------------------------------------------------
// K7: fused conv+BN+ReLU: out[b,o,pix] = relu(sA[o]*(W'x2)[o,pix] + sR[o]*spa[pix] + sB[o])
//     GEMM per b: M=64, K=64, N=65536 via V_WMMA_F32_16X16X4_F32.
//     Each wave: one (b, mtile); 16x64 A slice held in 32 VGPRs; streams 16
//     pixel-tiles (256 pixels) of x2 as B operands straight from global.
// ---------------------------------------------------------------------------
__global__ void k_fuse_wmma(const float* __restrict__ x2, const float* __restrict__ spa,
                            const float* __restrict__ Wp, const float* __restrict__ sA,
                            const float* __restrict__ sR, const float* __restrict__ sB,
                            float* __restrict__ out) {
    int tid = threadIdx.x;
    int lane = tid & 31, wv = tid >> 5;
    int wid = blockIdx.x * 8 + wv;     // 0..4095
    int b      = wid >> 10;            // 0..3
    int mtile  = (wid >> 8) & 3;       // 0..3
    int ngroup = wid & 255;            // 0..255 -> 256 pixels each
    int half = lane >> 4, ln = lane & 15;
    int m0 = mtile * 16;

    const float* arow = Wp + b * 4096 + (m0 + ln) * 64;
    v2f areg[16];
    #pragma unroll
    for (int k4 = 0; k4 < 16; ++k4) {
        int kk = k4 * 4 + (half ? 2 : 0);
        areg[k4].x = arow[kk];
        areg[k4].y = arow[kk + 1];
    }
    float cA[8], cR[8], cB[8];
    #pragma unroll
    for (int i = 0; i < 8; ++i) {
        int o = m0 + i + half * 8;
        cA[i] = sA[o]; cR[i] = sR[o]; cB[i] = sB[o];
    }
    const float* x2b  = x2  + (size_t)b * C2N * PIX2;
    const float* spab = spa + (size_t)b * PIX2;
    float* outb       = out + (size_t)b * C2N * PIX2;
    int pixbase = ngroup * 256;

    for (int nt = 0; nt < 16; ++nt) {
        int pix = pixbase + nt * 16 + ln;
        v8f c = {};
        #pragma unroll
        for (int k4 = 0; k4 < 16; ++k4) {
            int kch = k4 * 4 + (half ? 2 : 0);
            v2f bb;
            bb.x = x2b[(size_t)kch * PIX2 + pix];
            bb.y = x2b[(size_t)(kch + 1) * PIX2 + pix];
            c = __builtin_amdgcn_wmma_f32_16x16x4_f32(false, areg[k4], false, bb, (short)0, c, false, false);
        }
        float sv = spab[pix];
        #pragma unroll
        for (int i = 0; i < 8; ++i) {
            int o = m0 + i + half * 8;
            float v = c[i] * cA[i] + sv * cR[i] + cB[i];
            outb[(size_t)o * PIX2 + pix] = fmaxf(v, 0.f);
        }
    }
}

// ---------------------------------------------------------------------------
extern "C" void kernel_launch(void* const* d_in, const int* in_sizes, int n_in,
                              void* d_out, int out_size, void* d_ws, size_t ws_size,
                              hipStream_t stream) {
    const float* x1    = (const float*)d_in[0];
    const float* x2    = (const float*)d_in[1];
    const float* w_c1  = (const float*)d_in[2];
    const float* b_c1  = (const float*)d_in[3];
    const float* w_c2  = (const float*)d_in[4];
    const float* b_c2  = (const float*)d_in[5];
    const float* w_p1  = (const float*)d_in[6];
    const float* b_p1  = (const float*)d_in[7];
    const float* w_p2  = (const float*)d_in[8];
    const float* b_p2  = (const float*)d_in[9];
    const float* w_lin = (const float*)d_in[10];
    const float* b_lin = (const float*)d_in[11];
    const float* w_f   = (const float*)d_in[12];
    const float* b_f   = (const float*)d_in[13];
    const float* gamma = (const float*)d_in[14];
    const float* beta  = (const float*)d_in[15];
    float* out = (float*)d_out;

    float* ws = (float*)d_ws;
    float* gap1 = ws;                  // 512
    float* gap2 = gap1 + 512;          // 256
    float* cw   = gap2 + 256;          // 256
    float* pos1 = cw + 256;            // 65536
    float* pos2 = pos1 + 65536;        // 262144
    float* x1t  = pos2 + 262144;       // 262144
    float* spa  = x1t + 262144;        // 262144
    float* Wp   = spa + 262144;        // 16384
    float* sA   = Wp + 16384;          // 64
    float* sR   = sA + 64;             // 64
    float* sB   = sR + 64;             // 64

    k_plane_mean<<<BATCH * C1N + BATCH * C2N, 256, 0, stream>>>(x1, x2, gap1, gap2);
    k_pos1<<<BATCH * PIX1 / 256, 256, 0, stream>>>(x1, w_p1, b_p1, pos1);
    k_pos2<<<BATCH * PIX2 / 256, 256, 0, stream>>>(x2, w_p2, b_p2, pos2);
    k_channel_attn<<<1, 256, 0, stream>>>(gap1, gap2, w_c1, b_c1, w_c2, b_c2, cw);
    k_x1t_wmma<<<128, 256, 0, stream>>>(w_lin, b_lin, pos1, x1t);
    k_spa<<<BATCH * NB, 256, 0, stream>>>(pos2, x1t, spa);
    k_prep<<<64, 256, 0, stream>>>(w_f, b_f, gamma, beta, cw, Wp, sA, sR, sB);
    k_fuse_wmma<<<512, 256, 0, stream>>>(x2, spa, Wp, sA, sR, sB, out);
    (void)in_sizes; (void)n_in; (void)out_size; (void)ws_size;
}